// MoE_10136122818689
// MI455X (gfx1250) — compile-verified
//
#include <hip/hip_runtime.h>
#include <math.h>

// Problem constants (from reference): B=4, T=2048 -> NT=8192 tokens
#define NT   8192
#define D_   1024
#define E_   8

// GEMM tiling
#define TM   256     // rows per block (8 waves x 32)
#define TN   64      // cols per block (4 wmma col-tiles per wave)
#define KB   32      // K-chunk staged in LDS for B

// Workspace layout (bytes)
#define OFF_COUNTS   0
#define OFF_ENTRIES  256
#define OFF_W2       (256 + E_ * NT * 4)            // 256 + 256KB
#define OFF_SCRATCH  (OFF_W2 + NT * 2 * 4)          // + 64KB

typedef float v2f __attribute__((ext_vector_type(2)));
typedef float v8f __attribute__((ext_vector_type(8)));

__global__ void zero_counts(int* counts) {
    if (threadIdx.x < E_) counts[threadIdx.x] = 0;
}

// One wave32 per token: gate logits, softmax, top-2, route into expert lists.
__global__ void gate_kernel(const float* __restrict__ x,
                            const float* __restrict__ Wg,
                            int* __restrict__ counts,
                            int* __restrict__ entries,
                            float* __restrict__ w2) {
    const int wave = threadIdx.x >> 5;
    const int lane = threadIdx.x & 31;
    const int t = blockIdx.x * 8 + wave;
    if (t >= NT) return;

    const float* xr = x + (size_t)t * D_;
    float acc[E_];
#pragma unroll
    for (int e = 0; e < E_; ++e) acc[e] = 0.0f;

    for (int d = lane; d < D_; d += 32) {
        float xv = xr[d];
        float4 g0 = *(const float4*)(Wg + (size_t)d * E_);
        float4 g1 = *(const float4*)(Wg + (size_t)d * E_ + 4);
        acc[0] += xv * g0.x; acc[1] += xv * g0.y;
        acc[2] += xv * g0.z; acc[3] += xv * g0.w;
        acc[4] += xv * g1.x; acc[5] += xv * g1.y;
        acc[6] += xv * g1.z; acc[7] += xv * g1.w;
    }
    // wave32 butterfly reduction
#pragma unroll
    for (int off = 16; off >= 1; off >>= 1) {
#pragma unroll
        for (int e = 0; e < E_; ++e)
            acc[e] += __shfl_xor(acc[e], off, 32);
    }

    // top-2 of logits == top-2 of softmax; first-occurrence wins ties (matches top_k)
    int i0 = 0; float m0 = acc[0];
#pragma unroll
    for (int e = 1; e < E_; ++e) if (acc[e] > m0) { m0 = acc[e]; i0 = e; }
    int i1 = -1; float m1 = -3.4e38f;
#pragma unroll
    for (int e = 0; e < E_; ++e) if (e != i0 && acc[e] > m1) { m1 = acc[e]; i1 = e; }

    float Z = 0.0f;
#pragma unroll
    for (int e = 0; e < E_; ++e) Z += expf(acc[e] - m0);
    float v0 = 1.0f / Z;                 // exp(m0-m0)/Z
    float v1 = expf(m1 - m0) / Z;
    float s = v0 + v1 + 1e-9f;
    float w0 = v0 / s, w1 = v1 / s;

    if (lane == 0) {
        int p0 = atomicAdd(&counts[i0], 1);
        entries[i0 * NT + p0] = t;                       // kslot 0
        int p1 = atomicAdd(&counts[i1], 1);
        entries[i1 * NT + p1] = t | (int)0x80000000u;    // kslot 1
        w2[t * 2 + 0] = w0;
        w2[t * 2 + 1] = w1;
    }
}

// Grouped GEMM: for expert e, rows = routed tokens, y = x[rows] @ We[e][:, ncols]
// scratch[kslot][token][0..NW) gets the result (written exactly once -> deterministic).
// LDS B layout is K-pair interleaved: Bs2[k/2 * TN + n] = (B[k][n], B[k+1][n]) so each
// WMMA SRC1 operand is one aligned ds_load_b64 (no register shuffling).
__global__ void __launch_bounds__(256)
moe_gemm(const float* __restrict__ x,
         const float* __restrict__ We,
         const int* __restrict__ counts,
         const int* __restrict__ entries,
         float* __restrict__ scratch,
         int passOff, int NW) {
    __shared__ float Bs[KB * TN];   // 8KB, viewed as (KB/2) x TN float2
    v2f* Bs2 = (v2f*)Bs;

    const int e    = blockIdx.x >> 5;        // NT/TM = 32 row-tiles per expert
    const int mt   = blockIdx.x & 31;
    const int cnt  = counts[e];
    const int base = mt * TM;
    if (base >= cnt) return;

    const int wave = threadIdx.x >> 5;
    const int lane = threadIdx.x & 31;
    const int lm   = lane & 15;              // M (A) / N (B,C) index
    const int lh   = lane >> 4;              // K-half selector

    // Per-lane gathered A rows: wave owns 32 rows = two 16-row WMMA M-tiles
    const int rA0  = base + wave * 32 + lm;
    const int rA1  = rA0 + 16;
    const int eA0  = (rA0 < cnt) ? entries[e * NT + rA0] : 0;
    const int eA1  = (rA1 < cnt) ? entries[e * NT + rA1] : 0;
    const float* __restrict__ xrow0 = x + (size_t)(eA0 & 0x7fffffff) * D_;
    const float* __restrict__ xrow1 = x + (size_t)(eA1 & 0x7fffffff) * D_;

    const int ncol0 = passOff + blockIdx.y * TN;                 // global col base
    const float* __restrict__ Bp = We + (size_t)e * D_ * D_ + ncol0;

    v8f c[2][4];
#pragma unroll
    for (int m = 0; m < 2; ++m)
#pragma unroll
        for (int n = 0; n < 4; ++n) c[m][n] = (v8f){};

    for (int kc = 0; kc < D_; kc += KB) {
        // Stage B chunk K-pair interleaved: 256 threads, one 32B chunk each.
        {
            const int i  = threadIdx.x;          // chunk id: (KB/2)=16 kp-rows x 16 n-quads
            const int kp = i >> 4;
            const int n0 = (i & 15) * 4;
            float4 r0 = *(const float4*)(Bp + (size_t)(kc + 2 * kp + 0) * D_ + n0);
            float4 r1 = *(const float4*)(Bp + (size_t)(kc + 2 * kp + 1) * D_ + n0);
            float4* dst = (float4*)(Bs + 2 * (kp * TN + n0));
            dst[0] = make_float4(r0.x, r1.x, r0.y, r1.y);
            dst[1] = make_float4(r0.z, r1.z, r0.w, r1.w);
        }
        __syncthreads();

#pragma unroll
        for (int kk = 0; kk < KB; kk += 4) {
            const int kp = (kk >> 1) + lh;                 // float2 row for this lane
            v2f a0 = *(const v2f*)(xrow0 + kc + kk + 2 * lh);
            v2f a1 = *(const v2f*)(xrow1 + kc + kk + 2 * lh);
#pragma unroll
            for (int ct = 0; ct < 4; ++ct) {
                v2f b = Bs2[kp * TN + ct * 16 + lm];
                c[0][ct] = __builtin_amdgcn_wmma_f32_16x16x4_f32(false, a0, false, b, (short)0, c[0][ct], false, false);
                c[1][ct] = __builtin_amdgcn_wmma_f32_16x16x4_f32(false, a1, false, b, (short)0, c[1][ct], false, false);
            }
        }
        __syncthreads();
    }

    // Store: C layout -> element (m = v + 8*lh, n = lm) per VGPR v
    const int cloc = blockIdx.y * TN;   // col within pass [0, NW)
#pragma unroll
    for (int m = 0; m < 2; ++m) {
#pragma unroll
        for (int v = 0; v < 8; ++v) {
            const int r = base + wave * 32 + m * 16 + v + 8 * lh;
            if (r < cnt) {
                const int ent = entries[e * NT + r];
                const int tok = ent & 0x7fffffff;
                const int ks  = ((unsigned)ent) >> 31;
                float* orow = scratch + ((size_t)ks * NT + tok) * NW + cloc + lm;
                orow[ 0] = c[m][0][v];
                orow[16] = c[m][1][v];
                orow[32] = c[m][2][v];
                orow[48] = c[m][3][v];
            }
        }
    }
}

// out[t, passOff+j] = w0 * s0[t,j] + w1 * s1[t,j]
__global__ void combine(const float* __restrict__ scratch,
                        const float* __restrict__ w2,
                        float* __restrict__ out,
                        int passOff, int NW) {
    const int idx = blockIdx.x * 256 + threadIdx.x;     // float4 index
    const int total = NT * NW / 4;
    if (idx >= total) return;
    const int elem = idx * 4;
    const int t = elem / NW;
    const int j = elem % NW;
    const float w0 = w2[t * 2 + 0];
    const float w1 = w2[t * 2 + 1];
    float4 a = ((const float4*)scratch)[idx];
    float4 b = ((const float4*)scratch)[(size_t)(NT * (size_t)NW) / 4 + idx];
    float4 o;
    o.x = w0 * a.x + w1 * b.x;
    o.y = w0 * a.y + w1 * b.y;
    o.z = w0 * a.z + w1 * b.z;
    o.w = w0 * a.w + w1 * b.w;
    *(float4*)(out + (size_t)t * D_ + passOff + j) = o;
}

extern "C" void kernel_launch(void* const* d_in, const int* in_sizes, int n_in,
                              void* d_out, int out_size, void* d_ws, size_t ws_size,
                              hipStream_t stream) {
    const float* x  = (const float*)d_in[0];
    const float* Wg = (const float*)d_in[1];
    const float* We = (const float*)d_in[2];
    float* out = (float*)d_out;

    char* ws = (char*)d_ws;
    int*   counts  = (int*)(ws + OFF_COUNTS);
    int*   entries = (int*)(ws + OFF_ENTRIES);
    float* w2      = (float*)(ws + OFF_W2);
    float* scratch = (float*)(ws + OFF_SCRATCH);

    // Pick the fewest N-passes whose scratch (2 * NT * NW floats) fits the workspace.
    int p = 16;
    for (int cand = 1; cand <= 16; cand <<= 1) {
        size_t need = (size_t)OFF_SCRATCH + (size_t)2 * NT * (D_ / cand) * sizeof(float);
        if (need <= ws_size) { p = cand; break; }
    }
    const int NW = D_ / p;

    zero_counts<<<1, 64, 0, stream>>>(counts);
    gate_kernel<<<NT / 8, 256, 0, stream>>>(x, Wg, counts, entries, w2);

    for (int pass = 0; pass < p; ++pass) {
        dim3 grid(E_ * (NT / TM), NW / TN, 1);
        moe_gemm<<<grid, 256, 0, stream>>>(x, We, counts, entries, scratch,
                                           pass * NW, NW);
        const int total = NT * NW / 4;
        combine<<<(total + 255) / 256, 256, 0, stream>>>(scratch, w2, out,
                                                         pass * NW, NW);
    }
}